// det_nlx_53515292508349
// MI455X (gfx1250) — compile-verified
//
#include <hip/hip_runtime.h>
#include <hip/hip_bf16.h>

typedef float v2f __attribute__((ext_vector_type(2)));
typedef float v4f __attribute__((ext_vector_type(4)));
typedef float v8f __attribute__((ext_vector_type(8)));

#define NLOC 10
#define NGI  13
#define ROWS_PER_BLOCK 128   // 8 waves * 16 rows each
#define THREADS 256

// Per batch row b:
//   j11 = x . w11[g,:], j12 = y . w12[g,:], j21 = x . w21[g,:], j22 = y . w22[g,:]
//   det = j11*j22 - j21*j12 ; detwei = |det|*weight[g]
//   invj = adj(J)/det ; nx[d,loc,g] = invj[d][0]*nlx[0,loc,g] + invj[d][1]*nlx[1,loc,g]
//
// Phase 1 (per wave, 16 rows): WMMA f32 16x16x4 chains (K=12, zero-padded past 10)
//   compute j11/j12/j21/j22 tiles; elementwise det/invj in the C layout
//   (lane = g, VGPR index = M-row); scatter invj + detwei to LDS.
// Phase 2 (whole block): coalesced b128 non-temporal stores of nx
//   (128*260 floats, contiguous per block) and detwei (128*13 floats) from LDS.
__launch_bounds__(THREADS)
__global__ void det_nlx_wmma_kernel(const float* __restrict__ x_loc,
                                    const float* __restrict__ weight,
                                    const float* __restrict__ w11,
                                    const float* __restrict__ w12,
                                    const float* __restrict__ w21,
                                    const float* __restrict__ w22,
                                    const float* __restrict__ nlx,
                                    float* __restrict__ out_nx,
                                    float* __restrict__ out_detwei)
{
    __shared__ float lds_invj[ROWS_PER_BLOCK][4][NGI]; // [row][{11,12,21,22}][g]
    __shared__ float lds_dw[ROWS_PER_BLOCK][NGI];
    __shared__ float lds_nlx[2 * NLOC * NGI];          // nlx[d][loc][g]

    const int tid = threadIdx.x;

    // Stage the tiny nlx table (260 floats) into LDS.
    if (tid < 2 * NLOC * NGI) lds_nlx[tid] = nlx[tid];

    const int lane    = tid & 31;
    const int wave    = tid >> 5;
    const int g       = lane & 15;   // N index (g column) for B/C/D fragments
    const int hi      = lane >> 4;   // half-wave selector
    const int m       = lane & 15;   // M index (batch row) for A fragments
    const int rowBase = wave * 16;
    const long long bBase = (long long)blockIdx.x * ROWS_PER_BLOCK + rowBase;

    // ---- A fragments (branchless): clamp index so the address is always
    // legal, load unconditionally, then cndmask the value to zero for K>=10.
    // ISA f32 16x4 A layout: lanes 0-15 M=lane, VGPR0=K0,VGPR1=K1; lanes 16-31 K2,K3.
    const float* xrow = x_loc + (bBase + m) * (2 * NLOC);
    v2f ax[3], ay[3];
#pragma unroll
    for (int c = 0; c < 3; ++c) {
        const int k0  = 4 * c + 2 * hi;
        const int k0c = (k0     < NLOC) ? k0     : NLOC - 1;
        const int k1c = (k0 + 1 < NLOC) ? k0 + 1 : NLOC - 1;
        const float m0 = (k0     < NLOC) ? 1.f : 0.f;
        const float m1 = (k0 + 1 < NLOC) ? 1.f : 0.f;
        ax[c] = { xrow[k0c] * m0,        xrow[k1c] * m1 };
        ay[c] = { xrow[NLOC + k0c] * m0, xrow[NLOC + k1c] * m1 };
    }

    // ---- B fragments (branchless): B[k][n] = w[n][k] (w is [NGI][NLOC]);
    // zero-pad k>=10 and n>=13 via clamped in-bounds loads + value masking.
    // Mirror layout: lanes 0-15 N=lane, VGPR0=K0,VGPR1=K1; lanes 16-31 K2,K3.
    v2f b11[3], b12[3], b21[3], b22[3];
    const int gc = (g < NGI) ? g : NGI - 1;
#pragma unroll
    for (int c = 0; c < 3; ++c) {
        const int k0  = 4 * c + 2 * hi;
        const int k0c = (k0     < NLOC) ? k0     : NLOC - 1;
        const int k1c = (k0 + 1 < NLOC) ? k0 + 1 : NLOC - 1;
        const float m0 = (g < NGI && k0     < NLOC) ? 1.f : 0.f;
        const float m1 = (g < NGI && k0 + 1 < NLOC) ? 1.f : 0.f;
        const int i0 = gc * NLOC + k0c;
        const int i1 = gc * NLOC + k1c;
        b11[c] = { w11[i0] * m0, w11[i1] * m1 };
        b12[c] = { w12[i0] * m0, w12[i1] * m1 };
        b21[c] = { w21[i0] * m0, w21[i1] * m1 };
        b22[c] = { w22[i0] * m0, w22[i1] * m1 };
    }

    // ---- WMMA accumulation: J tiles, 3 chained K=4 steps each ----
    v8f c11 = {}, c12 = {}, c21 = {}, c22 = {};
#pragma unroll
    for (int c = 0; c < 3; ++c) {
        c11 = __builtin_amdgcn_wmma_f32_16x16x4_f32(false, ax[c], false, b11[c],
                                                    (short)0, c11, false, false);
        c12 = __builtin_amdgcn_wmma_f32_16x16x4_f32(false, ay[c], false, b12[c],
                                                    (short)0, c12, false, false);
        c21 = __builtin_amdgcn_wmma_f32_16x16x4_f32(false, ax[c], false, b21[c],
                                                    (short)0, c21, false, false);
        c22 = __builtin_amdgcn_wmma_f32_16x16x4_f32(false, ay[c], false, b22[c],
                                                    (short)0, c22, false, false);
    }

    // ---- Elementwise det / invj in the C layout; scatter to LDS ----
    const float wg = weight[gc];
#pragma unroll
    for (int r = 0; r < 8; ++r) {
        const float det = c11[r] * c22[r] - c21[r] * c12[r];
        const float inv = 1.0f / det;
        const int   row = rowBase + r + 8 * hi;  // C layout: lanes 0-15 M=r, 16-31 M=8+r
        if (g < NGI) {
            lds_invj[row][0][g] =  c22[r] * inv;   // invj11
            lds_invj[row][1][g] = -c12[r] * inv;   // invj12
            lds_invj[row][2][g] = -c21[r] * inv;   // invj21
            lds_invj[row][3][g] =  c11[r] * inv;   // invj22
            lds_dw[row][g]      = fabsf(det) * wg;
        }
    }

    __syncthreads();

    // ---- Phase 2: coalesced b128 non-temporal stores of this block's tile ----
    // nx region for rows [blockIdx*128, +128) is contiguous: 33280 floats = 8320 v4f,
    // and each block's base offset (133120 B) keeps 16-byte alignment.
    const long long nxBase = (long long)blockIdx.x * ROWS_PER_BLOCK * (2 * NLOC * NGI);
    constexpr int NX_VEC = ROWS_PER_BLOCK * 2 * NLOC * NGI / 4;   // 8320
#pragma unroll 2
    for (int vi = tid; vi < NX_VEC; vi += THREADS) {
        const int idx0 = vi * 4;
        v4f o;
#pragma unroll
        for (int e = 0; e < 4; ++e) {
            const int idx  = idx0 + e;
            const int row  = idx / 260;
            const int rem  = idx - row * 260;
            const int dd   = rem / 130;
            const int rem2 = rem - dd * 130;
            const int loc  = rem2 / NGI;
            const int gg   = rem2 - loc * NGI;
            o[e] = lds_invj[row][2 * dd][gg]     * lds_nlx[loc * NGI + gg]
                 + lds_invj[row][2 * dd + 1][gg] * lds_nlx[NLOC * NGI + loc * NGI + gg];
        }
        __builtin_nontemporal_store(o, (v4f*)(out_nx + nxBase + idx0));
    }

    // detwei region: 1664 floats = 416 v4f, block base offset 6656 B (16B aligned).
    const long long dwBase = (long long)blockIdx.x * ROWS_PER_BLOCK * NGI;
    constexpr int DW_VEC = ROWS_PER_BLOCK * NGI / 4;              // 416
    for (int vi = tid; vi < DW_VEC; vi += THREADS) {
        const int idx0 = vi * 4;
        v4f o;
#pragma unroll
        for (int e = 0; e < 4; ++e) {
            const int idx = idx0 + e;
            const int row = idx / NGI;
            const int gg  = idx - row * NGI;
            o[e] = lds_dw[row][gg];
        }
        __builtin_nontemporal_store(o, (v4f*)(out_detwei + dwBase + idx0));
    }
}

extern "C" void kernel_launch(void* const* d_in, const int* in_sizes, int n_in,
                              void* d_out, int out_size, void* d_ws, size_t ws_size,
                              hipStream_t stream) {
    const float* x_loc  = (const float*)d_in[0];
    const float* weight = (const float*)d_in[1];
    const float* w11    = (const float*)d_in[2];
    const float* w12    = (const float*)d_in[3];
    const float* w21    = (const float*)d_in[4];
    const float* w22    = (const float*)d_in[5];
    const float* nlx    = (const float*)d_in[6];

    const int B = in_sizes[0] / (2 * NLOC);            // 524288
    float* out_nx     = (float*)d_out;                 // (B,2,10,13) flat
    float* out_detwei = (float*)d_out + (size_t)B * (2 * NLOC * NGI);

    const int grid = B / ROWS_PER_BLOCK;               // 4096 blocks
    det_nlx_wmma_kernel<<<grid, THREADS, 0, stream>>>(
        x_loc, weight, w11, w12, w21, w22, nlx, out_nx, out_detwei);
}